// LaneATTNMS_12936441495675
// MI455X (gfx1250) — compile-verified
//
#include <hip/hip_runtime.h>
#include <hip/hip_bf16.h>
#include <math.h>

#define N_OFFSETS 72
#define N_STRIPS  71
#define PROPSZ    77
#define NPROP     4096
#define BATCH     32
#define TOPK      5
#define NMS_THRES 50.0f
#define CONF_TH   0.4f
#define NTHREADS  1024
#define NWAVES    32
#define NTILES    (NPROP / 16)   // 256

typedef float v2f __attribute__((ext_vector_type(2)));
typedef float v8f __attribute__((ext_vector_type(8)));

__global__ __launch_bounds__(NTHREADS)
void laneatt_nms_gfx1250(const float* __restrict__ props,    // [B,N,77]
                         const float* __restrict__ anchors,  // [N,77]
                         const float* __restrict__ attn,     // [B,N,77]
                         float* __restrict__ out)
{
    __shared__ float s_score[NPROP];          // -inf == inactive
    __shared__ int   s_se[NPROP];             // starts | (ends<<16), ends signed
    __shared__ float s_cxs[N_OFFSETS];        // chosen proposal's xs
    __shared__ float s_rv[NWAVES];
    __shared__ int   s_ri[NWAVES];
    __shared__ int   s_keep[TOPK];
    __shared__ float s_vm[TOPK];
    __shared__ int   s_bidx;
    __shared__ float s_bval;
    __shared__ int   s_bse;

    const int b     = blockIdx.x;
    const int tid   = threadIdx.x;
    const int lane  = tid & 31;
    const int wave  = tid >> 5;
    const int khalf = (lane >> 4) * 2;        // 0 or 2 (K half this lane owns)
    const float* P = props + (size_t)b * NPROP * PROPSZ;
    const float* A = attn  + (size_t)b * NPROP * PROPSZ;

    // ---------- Phase A: scores + packed (start,end) ----------
    for (int i = tid; i < NPROP; i += NTHREADS) {
        const float* r = P + (size_t)i * PROPSZ;
        float l0 = r[0], l1 = r[1];
        float score = 1.0f / (1.0f + expf(l0 - l1));      // softmax[...,1]
        s_score[i] = (score > CONF_TH) ? score : -INFINITY;
        int st  = (int)rintf(r[2] * (float)N_STRIPS);     // jnp.round -> half-even
        int len = (int)r[4];                              // astype(int32) truncation
        int en  = min(st + len - 1, N_STRIPS);            // may be -1
        s_se[i] = (st & 0xffff) | (en << 16);
    }
    __syncthreads();

    // ---------- Phase B: greedy NMS, TOPK rounds ----------
    for (int it = 0; it < TOPK; ++it) {
        // block argmax (lowest index wins ties, matches jnp.argmax)
        float bv = -INFINITY; int bi = 0x7fffffff;
        for (int i = tid; i < NPROP; i += NTHREADS) {
            float v = s_score[i];
            if (v > bv || (v == bv && i < bi)) { bv = v; bi = i; }
        }
        #pragma unroll
        for (int off = 16; off >= 1; off >>= 1) {
            float ov = __shfl_xor(bv, off, 32);
            int   oi = __shfl_xor(bi, off, 32);
            if (ov > bv || (ov == bv && oi < bi)) { bv = ov; bi = oi; }
        }
        if (lane == 0) { s_rv[wave] = bv; s_ri[wave] = bi; }
        __syncthreads();
        if (wave == 0) {
            bv = s_rv[lane]; bi = s_ri[lane];
            #pragma unroll
            for (int off = 16; off >= 1; off >>= 1) {
                float ov = __shfl_xor(bv, off, 32);
                int   oi = __shfl_xor(bi, off, 32);
                if (ov > bv || (ov == bv && oi < bi)) { bv = ov; bi = oi; }
            }
            if (lane == 0) {
                bool valid = (bv > -INFINITY);
                int  kidx  = valid ? bi : 0;
                s_bidx = kidx;
                s_bval = bv;
                s_bse  = s_se[kidx];
                s_keep[it] = kidx;
                s_vm[it]   = valid ? 1.0f : 0.0f;
            }
        }
        __syncthreads();

        const bool valid = (s_bval > -INFINITY);   // block-uniform
        if (valid) {
            const int bidx = s_bidx;
            if (tid < N_OFFSETS)
                s_cxs[tid] = P[(size_t)bidx * PROPSZ + 5 + tid];
            __syncthreads();

            const int se_i = s_bse;
            const int si = se_i & 0xffff;
            const int ei = se_i >> 16;             // arithmetic shift keeps -1

            // Hoist this lane's 36 chosen-row xs values into registers:
            // invariant across all tiles this wave processes this round.
            float cxs_l[36];
            #pragma unroll
            for (int c = 0; c < 18; ++c) {
                cxs_l[2*c]   = s_cxs[c*4 + khalf];
                cxs_l[2*c+1] = s_cxs[c*4 + khalf + 1];
            }

            // Suppression: each wave owns 8 tiles of 16 proposals.
            // dist row-sums via V_WMMA_F32_16X16X4_F32 with B = ones
            // (result invariant to A's K-slot layout; only M<->lane matters).
            for (int tile = wave; tile < NTILES; tile += NWAVES) {
                const int base = tile * 16;
                const int row  = base + (lane & 15);
                const float* xr = P + (size_t)row * PROPSZ + 5;

                // prefetch next tile's row for this lane group
                const int ntile = tile + NWAVES;
                if (ntile < NTILES)
                    __builtin_prefetch(P + (size_t)(ntile * 16 + (lane & 15)) * PROPSZ + 5, 0, 3);

                const int sej = s_se[row];
                const int lo = max(sej & 0xffff, si);
                const int hi = min(sej >> 16,    ei);

                // Phase 1: issue ALL row loads unconditionally (deep MLP,
                // one wait instead of 36 serialized load+wait pairs).
                float xv[36];
                #pragma unroll
                for (int c = 0; c < 18; ++c) {
                    xv[2*c]   = xr[c*4 + khalf];
                    xv[2*c+1] = xr[c*4 + khalf + 1];
                }

                // Phase 2: masked |diff| (branchless cndmask) + WMMA chain.
                v2f bones; bones.x = 1.0f; bones.y = 1.0f;
                v8f acc = {0.f,0.f,0.f,0.f,0.f,0.f,0.f,0.f};
                #pragma unroll
                for (int c = 0; c < 18; ++c) {
                    const int t0 = c*4 + khalf;
                    const int t1 = t0 + 1;
                    const float d0 = fabsf(xv[2*c]   - cxs_l[2*c]);
                    const float d1 = fabsf(xv[2*c+1] - cxs_l[2*c+1]);
                    v2f a;
                    a.x = (t0 >= lo && t0 <= hi) ? d0 : 0.0f;
                    a.y = (t1 >= lo && t1 <= hi) ? d1 : 0.0f;
                    acc = __builtin_amdgcn_wmma_f32_16x16x4_f32(
                        /*neg_a=*/false, a, /*neg_b=*/false, bones,
                        /*c_mod=*/(short)0, acc, /*reuse_a=*/false, /*reuse_b=*/false);
                }

                // lane 0 holds D rows 0..7, lane 16 holds rows 8..15 (col 0)
                if ((lane & 15) == 0) {
                    const int rbase = base + ((lane >> 4) << 3);
                    #pragma unroll
                    for (int r = 0; r < 8; ++r) {
                        const int j   = rbase + r;
                        const int sj  = s_se[j];
                        const int l2  = max(sj & 0xffff, si);
                        const int h2  = min(sj >> 16,    ei);
                        const float thr = NMS_THRES * (float)(h2 - l2 + 1);
                        if (acc[r] < thr) s_score[j] = -INFINITY;
                    }
                }
            }
            __syncthreads();
        }
    }
    __syncthreads();

    // ---------- Phase C: masked gather ----------
    float* o_prop = out;
    float* o_anch = out + (size_t)BATCH * TOPK * PROPSZ;
    float* o_attn = out + (size_t)2 * BATCH * TOPK * PROPSZ;
    float* o_keep = out + (size_t)3 * BATCH * TOPK * PROPSZ;
    float* o_vm   = o_keep + BATCH * TOPK;

    for (int idx = tid; idx < TOPK * PROPSZ; idx += NTHREADS) {
        const int k  = idx / PROPSZ;
        const int c  = idx % PROPSZ;
        const int j  = s_keep[k];
        const float m = s_vm[k];
        const size_t ob = (size_t)b * TOPK * PROPSZ + idx;
        o_prop[ob] = P[(size_t)j * PROPSZ + c] * m;
        o_anch[ob] = anchors[(size_t)j * PROPSZ + c] * m;
        o_attn[ob] = A[(size_t)j * PROPSZ + c] * m;
    }
    if (tid < TOPK) {
        o_keep[b * TOPK + tid] = (float)s_keep[tid];
        o_vm[b * TOPK + tid]   = s_vm[tid];
    }
}

extern "C" void kernel_launch(void* const* d_in, const int* in_sizes, int n_in,
                              void* d_out, int out_size, void* d_ws, size_t ws_size,
                              hipStream_t stream) {
    const float* props   = (const float*)d_in[0];   // (32,4096,77)
    const float* anchors = (const float*)d_in[1];   // (4096,77)
    const float* attn    = (const float*)d_in[2];   // (32,4096,77)
    float* out = (float*)d_out;
    (void)in_sizes; (void)n_in; (void)out_size; (void)d_ws; (void)ws_size;
    laneatt_nms_gfx1250<<<BATCH, NTHREADS, 0, stream>>>(props, anchors, attn, out);
}